// TropicalMatrixMul_46557445489434
// MI455X (gfx1250) — compile-verified
//
#include <hip/hip_runtime.h>
#include <stdint.h>

// Tropical (min-plus) matmul: out[b,o] = min_j (x[b,j] + w[o,j])
// x: [B, IN] f32 row-major, w: [OUT, IN] f32 row-major, out: [B, OUT] f32.
#define B_DIM   512
#define IN_DIM  512
#define OUT_DIM 1024

#define TM 64          // batch tile
#define TN 64          // out-feature tile
#define TK 32          // k chunk
#define LDS_PITCH 36   // floats per LDS tile row: 144B = 9*16B (16B-aligned rows, bank-rotating)

// ---- gfx1250 async global->LDS copy (ASYNCcnt path) ----
__device__ __forceinline__ void async_copy_b128(uint32_t lds_byte_off, const float* gptr) {
    // VDST = VGPR holding LDS byte address, VADDR = 64-bit global address, GV mode.
    asm volatile("global_load_async_to_lds_b128 %0, %1, off"
                 :: "v"(lds_byte_off), "v"(gptr)
                 : "memory");
}

// Wait until at most N async ops remain outstanding (async loads complete in order,
// so waiting <=4 after issuing the next tile's 4 copies guarantees the previous
// tile's 4 copies have landed in LDS).
#define WAIT_ASYNC_LE(n) asm volatile("s_wait_asynccnt " #n ::: "memory")

__global__ __launch_bounds__(256)
void tropical_minplus_kernel(const float* __restrict__ x,
                             const float* __restrict__ w,
                             float* __restrict__ out) {
    // Double-buffered tiles: 2 * 2 * 64*36*4B = 36864 B of LDS (WGP has 320 KB).
    __shared__ float xs[2][TM * LDS_PITCH];
    __shared__ float ws[2][TN * LDS_PITCH];

    const int tid = threadIdx.x;     // 0..255
    const int tx  = tid & 15;        // out-col group (0..15) -> 4 cols each
    const int ty  = tid >> 4;        // batch-row group (0..15) -> 4 rows each

    const int block_m = blockIdx.y * TM;   // batch base
    const int block_n = blockIdx.x * TN;   // out-feature base

    float acc[4][4];
#pragma unroll
    for (int i = 0; i < 4; ++i)
#pragma unroll
        for (int j = 0; j < 4; ++j)
            acc[i][j] = __builtin_inff();  // identity of min

    // Copy assignment: tile is 64 rows x 8 float4-quads = 512 xfers per matrix;
    // 256 threads -> 2 per thread per matrix.
    const int id0  = tid;                  // 0..255
    const int id1  = tid + 256;            // 256..511
    const int row0 = id0 >> 3, kq0 = id0 & 7;
    const int row1 = id1 >> 3, kq1 = id1 & 7;

    uint32_t xs_off0[2], xs_off1[2], ws_off0[2], ws_off1[2];
#pragma unroll
    for (int b = 0; b < 2; ++b) {
        xs_off0[b] = (uint32_t)(uintptr_t)&xs[b][row0 * LDS_PITCH + kq0 * 4];
        xs_off1[b] = (uint32_t)(uintptr_t)&xs[b][row1 * LDS_PITCH + kq1 * 4];
        ws_off0[b] = (uint32_t)(uintptr_t)&ws[b][row0 * LDS_PITCH + kq0 * 4];
        ws_off1[b] = (uint32_t)(uintptr_t)&ws[b][row1 * LDS_PITCH + kq1 * 4];
    }

    const float* gx0 = x + (size_t)(block_m + row0) * IN_DIM + kq0 * 4;
    const float* gx1 = x + (size_t)(block_m + row1) * IN_DIM + kq1 * 4;
    const float* gw0 = w + (size_t)(block_n + row0) * IN_DIM + kq0 * 4;
    const float* gw1 = w + (size_t)(block_n + row1) * IN_DIM + kq1 * 4;

    // Preload tile 0 into buffer 0.
    async_copy_b128(xs_off0[0], gx0);
    async_copy_b128(xs_off1[0], gx1);
    async_copy_b128(ws_off0[0], gw0);
    async_copy_b128(ws_off1[0], gw1);

    int buf = 0;
    for (int k0 = 0; k0 < IN_DIM; k0 += TK) {
        const int knext = k0 + TK;
        if (knext < IN_DIM) {
            // Prefetch next tile into the other buffer (safe: all waves passed
            // the barrier that ended the iteration which last read it).
            const int nb = buf ^ 1;
            async_copy_b128(xs_off0[nb], gx0 + knext);
            async_copy_b128(xs_off1[nb], gx1 + knext);
            async_copy_b128(ws_off0[nb], gw0 + knext);
            async_copy_b128(ws_off1[nb], gw1 + knext);
            WAIT_ASYNC_LE(0x4);   // previous tile's 4 copies are done (in-order)
        } else {
            WAIT_ASYNC_LE(0x0);
        }
        __syncthreads();          // all waves' copies for `buf` visible

        const float* __restrict__ xt = xs[buf];
        const float* __restrict__ wt = ws[buf];

#pragma unroll
        for (int kk = 0; kk < TK; kk += 4) {
            float4 a4[4], b4[4];
#pragma unroll
            for (int i = 0; i < 4; ++i)
                a4[i] = *(const float4*)&xt[(ty * 4 + i) * LDS_PITCH + kk];
#pragma unroll
            for (int j = 0; j < 4; ++j)
                b4[j] = *(const float4*)&wt[(tx * 4 + j) * LDS_PITCH + kk];

#pragma unroll
            for (int i = 0; i < 4; ++i) {
#pragma unroll
                for (int j = 0; j < 4; ++j) {
                    // Tree reduction: depth 2, foldable into v_min3_num_f32.
                    float s0 = a4[i].x + b4[j].x;
                    float s1 = a4[i].y + b4[j].y;
                    float s2 = a4[i].z + b4[j].z;
                    float s3 = a4[i].w + b4[j].w;
                    float m01 = fminf(s0, s1);
                    float m23 = fminf(s2, s3);
                    acc[i][j] = fminf(acc[i][j], fminf(m01, m23));
                }
            }
        }
        __syncthreads();          // safe to overwrite `buf` next time around
        buf ^= 1;
    }

    // Write out: 4 consecutive out-columns per thread -> float4 stores.
#pragma unroll
    for (int i = 0; i < 4; ++i) {
        const int b = block_m + ty * 4 + i;
        const int o = block_n + tx * 4;
        float4 v = make_float4(acc[i][0], acc[i][1], acc[i][2], acc[i][3]);
        *(float4*)&out[(size_t)b * OUT_DIM + o] = v;
    }
}

extern "C" void kernel_launch(void* const* d_in, const int* in_sizes, int n_in,
                              void* d_out, int out_size, void* d_ws, size_t ws_size,
                              hipStream_t stream) {
    const float* x = (const float*)d_in[0];   // [B, IN]
    const float* w = (const float*)d_in[1];   // [OUT, IN]
    float* out     = (float*)d_out;           // [B, OUT]

    dim3 grid(OUT_DIM / TN, B_DIM / TM);      // 16 x 8 = 128 workgroups
    dim3 block(256);
    tropical_minplus_kernel<<<grid, block, 0, stream>>>(x, w, out);
}